// yoloLoss_66340064854039
// MI455X (gfx1250) — compile-verified
//
#include <hip/hip_runtime.h>
#include <math.h>

// ---------------------------------------------------------------------------
// YOLO loss on MI455X (gfx1250). Bandwidth-bound: 141 MB -> 1 float.
// Data path: async global->LDS staging (GLOBAL_LOAD_ASYNC_TO_LDS_B64,
// ASYNCcnt) per 512-cell tile, compute from LDS, then a deterministic
// reduction tree whose wave-level sum runs on the matrix pipe
// (V_WMMA_F32_16X16X4_F32 with B = ones) + ds_swizzle SWAPX16.
// ---------------------------------------------------------------------------

typedef float v2f __attribute__((ext_vector_type(2)));
typedef float v8f __attribute__((ext_vector_type(8)));
typedef int   v2i __attribute__((ext_vector_type(2)));

// address-space-qualified element typedefs for the async-LDS builtin
typedef __attribute__((address_space(1))) v2i glob_v2i;   // global memory
typedef __attribute__((address_space(3))) v2i lds_v2i;    // LDS

#define CELL_SZ (448.0f / 7.0f)
#define IMG_SZ  448.0f

#define THREADS     256
#define TILE_CELLS  512                       // cells per block tile
#define TILE_FLOATS (TILE_CELLS * 11)         // 5632 floats = 22528 B
#define TILE_VEC2   (TILE_FLOATS / 2)         // 2816 x b64 chunks
#define VEC2_PER_T  (TILE_VEC2 / THREADS)     // 11 b64 loads per thread/tensor
#define CELLS_PER_T (TILE_CELLS / THREADS)    // 2 cells per thread

#if __has_builtin(__builtin_amdgcn_global_load_async_to_lds_b64)
#define HAVE_ASYNC_LDS 1
#else
#define HAVE_ASYNC_LDS 0
#endif

// ---------------- wave / block reduction (WMMA on the matrix pipe) ---------
__device__ __forceinline__ float wave_sum_wmma(float v) {
    v2f a; a[0] = v;    a[1] = 0.0f;     // each lane's partial appears once in A
    v2f b; b[0] = 1.0f; b[1] = 1.0f;     // B = all-ones (4x16)
    v8f c = {};
    // D = A x B : every column of D = sum of all 64 A elements = wave total
    v8f d = __builtin_amdgcn_wmma_f32_16x16x4_f32(
        false, a, false, b, (short)0, c, false, false);
    float s = ((d[0] + d[1]) + (d[2] + d[3])) + ((d[4] + d[5]) + (d[6] + d[7]));
    // other half of the column lives in lane L^16 (SWAPX16 swizzle)
    int si = __builtin_amdgcn_ds_swizzle(__float_as_int(s), 0x401F);
    s += __int_as_float(si);
    return s; // wave total in every lane
}

__device__ __forceinline__ float block_sum_256(float v) {
    __shared__ float wsum[8];
    float w = wave_sum_wmma(v);
    if ((threadIdx.x & 31) == 0) wsum[threadIdx.x >> 5] = w;
    __syncthreads();
    float total = 0.0f;
    if (threadIdx.x == 0) {
        #pragma unroll
        for (int i = 0; i < 8; ++i) total += wsum[i]; // fixed order
    }
    return total; // valid in thread 0 only
}

// ---------------- per-cell YOLO loss ---------------------------------------
__device__ __forceinline__ float cell_loss(const float* __restrict__ p,
                                           const float* __restrict__ t) {
    float dc = p[10] - t[10];
    float loss = dc * dc;                       // class loss (always)

    float t4 = t[4];
    if (t4 > 0.0f) {
        // target box (channels 0..3) -> xyxy pixels
        float tcx = t[0] * CELL_SZ, tcy = t[1] * CELL_SZ;
        float tw  = t[2] * IMG_SZ,  th  = t[3] * IMG_SZ;
        float tx1 = tcx - 0.5f * tw, ty1 = tcy - 0.5f * th;
        float tx2 = tcx + 0.5f * tw, ty2 = tcy + 0.5f * th;
        float area_t = tw * th;

        float iou0 = 0.0f, iou1 = 0.0f;
        #pragma unroll
        for (int j = 0; j < 2; ++j) {
            float cx = p[5*j + 0] * CELL_SZ, cy = p[5*j + 1] * CELL_SZ;
            float w  = p[5*j + 2] * IMG_SZ,  h  = p[5*j + 3] * IMG_SZ;
            float x1 = cx - 0.5f * w, y1 = cy - 0.5f * h;
            float x2 = cx + 0.5f * w, y2 = cy + 0.5f * h;
            float ltx = fmaxf(x1, tx1), lty = fmaxf(y1, ty1);
            float rbx = fminf(x2, tx2), rby = fminf(y2, ty2);
            float iou = 0.0f;
            if ((ltx < rbx) && (lty < rby)) {
                float inter = (rbx - ltx) * (rby - lty);
                iou = inter / (w * h + area_t - inter);
            }
            if (j == 0) iou0 = iou; else iou1 = iou;
        }
        int   r      = (iou0 > iou1) ? 0 : 1;
        float maxiou = fmaxf(iou0, iou1);
        float pc_r   = p[5*r + 4];
        float pc_o   = p[9 - 5*r];

        float dconf = pc_r - maxiou;
        loss += dconf * dconf;                  // contain
        loss += 0.5f * pc_o * pc_o;             // 0.5 * not_contain

        const float* pp = p + 5*r;              // loc vs target pair slot r
        const float* tp = t + 5*r;
        float dx = pp[0] - tp[0];
        float dy = pp[1] - tp[1];
        float dw = sqrtf(pp[2]) - sqrtf(tp[2]);
        float dh = sqrtf(pp[3]) - sqrtf(tp[3]);
        loss += 5.0f * ((dx*dx + dy*dy) + (dw*dw + dh*dh));  // L_COORD
    } else {
        float a = p[4] - t4;
        float b = p[9] - t[9];
        loss += 0.5f * (a*a + b*b);             // L_NOOBJ
    }
    return loss;
}

// ---------------- kernel 1: one 512-cell tile per block --------------------
__global__ __launch_bounds__(THREADS)
void yolo_loss_partial(const float* __restrict__ pred,
                       const float* __restrict__ tgt,
                       float* __restrict__ partials) {
    __shared__ float tile[2 * TILE_FLOATS];     // 45056 B LDS
    float* tp_ = tile;                          // pred tile
    float* tt_ = tile + TILE_FLOATS;            // tgt  tile
    const size_t tileBase = (size_t)blockIdx.x * TILE_FLOATS;
    const int tid = threadIdx.x;

#if HAVE_ASYNC_LDS
    #pragma unroll
    for (int k = 0; k < VEC2_PER_T; ++k) {
        int v = tid + k * THREADS;              // lane-contiguous 8B chunks
        __builtin_amdgcn_global_load_async_to_lds_b64(
            (glob_v2i*)(pred + tileBase + 2*v),
            (lds_v2i*)(tp_ + 2*v), 0, 0);
        __builtin_amdgcn_global_load_async_to_lds_b64(
            (glob_v2i*)(tgt + tileBase + 2*v),
            (lds_v2i*)(tt_ + 2*v), 0, 0);
    }
#if __has_builtin(__builtin_amdgcn_s_wait_asynccnt)
    __builtin_amdgcn_s_wait_asynccnt(0);
#else
    asm volatile("s_wait_asynccnt 0x0" ::: "memory");
#endif
#else
    // fallback: stage through VGPRs with 8B vector copies
    #pragma unroll
    for (int k = 0; k < VEC2_PER_T; ++k) {
        int v = tid + k * THREADS;
        *(float2*)(tp_ + 2*v) = *(const float2*)(pred + tileBase + 2*v);
        *(float2*)(tt_ + 2*v) = *(const float2*)(tgt  + tileBase + 2*v);
    }
#endif
    __syncthreads();                            // all waves' tiles visible

    float acc = 0.0f;
    #pragma unroll
    for (int c = 0; c < CELLS_PER_T; ++c) {
        int cell = tid * CELLS_PER_T + c;       // 22-dword lane stride: no bank conflicts
        acc += cell_loss(tp_ + cell * 11, tt_ + cell * 11);
    }

    float bsum = block_sum_256(acc);
    if (tid == 0) partials[blockIdx.x] = bsum;
}

// ---------------- kernel 2: deterministic finish ---------------------------
__global__ __launch_bounds__(THREADS)
void yolo_loss_final(const float* __restrict__ partials,
                     float* __restrict__ out, int nblk, float invN) {
    float acc = 0.0f;
    for (int i = threadIdx.x; i < nblk; i += THREADS) acc += partials[i];
    float tot = block_sum_256(acc);
    if (threadIdx.x == 0) out[0] = tot * invN;
}

extern "C" void kernel_launch(void* const* d_in, const int* in_sizes, int n_in,
                              void* d_out, int out_size, void* d_ws, size_t ws_size,
                              hipStream_t stream) {
    const float* pred = (const float*)d_in[0];
    const float* tgt  = (const float*)d_in[1];
    float* out        = (float*)d_out;
    float* partials   = (float*)d_ws;           // nblk floats of scratch

    const int ncells = in_sizes[0] / 11;        // N * 7 * 7 = 1,605,632
    const int N      = ncells / 49;
    const int nblk   = ncells / TILE_CELLS;     // 3136 tiles, exact

    yolo_loss_partial<<<nblk, THREADS, 0, stream>>>(pred, tgt, partials);
    yolo_loss_final<<<1, THREADS, 0, stream>>>(partials, out, nblk, 1.0f / (float)N);
}